// GraphSAGE_11038065951061
// MI455X (gfx1250) — compile-verified
//
#include <hip/hip_runtime.h>

// CDNA5 (gfx1250) fused GraphSAGE layer: gather+mean -> [x|agg] @ W + b -> relu
// using V_WMMA_F32_16X16X4_F32 (full fp32 matrix path, wave32).

typedef __attribute__((ext_vector_type(2))) float v2f;
typedef __attribute__((ext_vector_type(8))) float v8f;

#define DDIM   128     // feature dim (fixed by problem)
#define ROWS   64      // rows (nodes) per block
#define KTOT   256     // 2*DDIM
#define KCHUNK 64      // W rows staged in LDS per chunk
#define NCHNK  4       // KTOT / KCHUNK

// Swizzled A-tile:  [rTile 0..3][kfrag 0..63][lane 0..31][2]   -> 16384 floats (64KB)
// Swizzled W-chunk: [nTile 0..7][kfrag 0..15][lane 0..31][2]   ->  8192 floats (32KB)
#define A_FLOATS 16384
#define LDS_FLOATS (A_FLOATS + 8192)

// A fragment (16x4, M x K), per ISA layout:
//   lane<16 : {A[M=lane][4k+0], A[M=lane][4k+1]}
//   lane>=16: {A[M=lane-16][4k+2], A[M=lane-16][4k+3]}
// B fragment (4x16, K x N):
//   lane<16 : {B[4k+0][n=lane], B[4k+1][n=lane]}
//   lane>=16: {B[4k+2][n=lane-16], B[4k+3][n=lane-16]}
__device__ __forceinline__ int a_swizzle(int row, int col) {
    int r    = row >> 4;
    int kf   = col >> 2;
    int lane = (row & 15) + (((col >> 1) & 1) << 4);
    int sub  = col & 1;
    return (((r << 6) + kf) << 6) + (lane << 1) + sub;
}
__device__ __forceinline__ int w_swizzle(int kk, int col) {
    int n    = col >> 4;
    int kf   = kk >> 2;
    int lane = (col & 15) + (((kk >> 1) & 1) << 4);
    int sub  = kk & 1;
    return A_FLOATS + (((n << 4) + kf) << 6) + (lane << 1) + sub;
}

__global__ __launch_bounds__(256)
void sage_layer_wmma(const float* __restrict__ xin,
                     const int*   __restrict__ nbr,
                     const float* __restrict__ W,
                     const float* __restrict__ bias,
                     float*       __restrict__ out,
                     int N, int deg)
{
    __shared__ float lds[LDS_FLOATS];   // 96KB, fine on CDNA5 (<=320KB/WG)

    const int t     = threadIdx.x;
    const int lane  = t & 31;
    const int wv    = t >> 5;       // 8 waves
    const int rTile = wv & 3;       // 4 row tiles of 16
    const int ch    = wv >> 2;      // 2 column halves (4 n-tiles each)
    const int base  = blockIdx.x * ROWS;

    // ---- Phase A1: self features  A[row][0..127] = x[row][:] -------------
    for (int i = t; i < ROWS * DDIM; i += 256) {
        int row = i >> 7, col = i & 127;
        int g = base + row;
        float v = (g < N) ? xin[g * DDIM + col] : 0.0f;
        lds[a_swizzle(row, col)] = v;
    }

    // ---- Phase A2: neighbor mean  A[row][128..255] -----------------------
    {
        const int grp  = t >> 7;    // 0..1 (two 128-thread groups)
        const int c128 = t & 127;
        const float inv = 1.0f / (float)deg;
        for (int row = grp; row < ROWS; row += 2) {
            int g = base + row;
            float acc = 0.0f;
            if (g < N) {
                const int* np = nbr + g * deg;
                for (int j = 0; j < deg; ++j) {
                    int nb = np[j];
                    acc += xin[nb * DDIM + c128];   // coalesced 512B row read (L2 hot)
                }
                acc *= inv;
            }
            lds[a_swizzle(row, DDIM + c128)] = acc;
        }
    }
    __syncthreads();

    // ---- Phase B: GEMM over K=256 in 4 LDS-staged W chunks ---------------
    v8f acc[4] = {v8f{}, v8f{}, v8f{}, v8f{}};

    for (int cc = 0; cc < NCHNK; ++cc) {
        // stage W rows [cc*64, cc*64+64) into swizzled LDS
        for (int i = t; i < KCHUNK * DDIM; i += 256) {
            int kk = i >> 7, c = i & 127;
            lds[w_swizzle(kk, c)] = W[(cc * KCHUNK + kk) * DDIM + c];
        }
        __syncthreads();

#pragma unroll
        for (int k = 0; k < 16; ++k) {
            v2f a = *(const v2f*)&lds[(((rTile << 6) + (cc << 4) + k) << 6) + (lane << 1)];
            int bb = A_FLOATS + (k << 6) + (lane << 1);
#pragma unroll
            for (int t4 = 0; t4 < 4; ++t4) {
                int n = (ch << 2) + t4;
                v2f b = *(const v2f*)&lds[bb + (n << 10)];
                acc[t4] = __builtin_amdgcn_wmma_f32_16x16x4_f32(
                    false, a, false, b, (short)0, acc[t4], false, false);
            }
        }
        __syncthreads();
    }

    // ---- Epilogue: bias + relu + store -----------------------------------
    // D layout: VGPR e -> M = e + 8*(lane>=16), N = lane%16
    const int nlo    = lane & 15;
    const int rowOff = base + (rTile << 4) + ((lane >> 4) << 3);
#pragma unroll
    for (int t4 = 0; t4 < 4; ++t4) {
        int col = ((ch << 2) + t4) * 16 + nlo;
        float bv = bias[col];
#pragma unroll
        for (int e = 0; e < 8; ++e) {
            int row = rowOff + e;
            if (row < N) {
                float v = acc[t4][e] + bv;
                out[row * DDIM + col] = v > 0.0f ? v : 0.0f;
            }
        }
    }
}

extern "C" void kernel_launch(void* const* d_in, const int* in_sizes, int n_in,
                              void* d_out, int out_size, void* d_ws, size_t ws_size,
                              hipStream_t stream) {
    const float* x   = (const float*)d_in[0];
    const int*   nbr = (const int*)  d_in[1];
    const float* W1  = (const float*)d_in[2];
    const float* b1  = (const float*)d_in[3];
    const float* W2  = (const float*)d_in[4];
    const float* b2  = (const float*)d_in[5];
    float* out = (float*)d_out;

    const int N   = in_sizes[0] / DDIM;
    const int deg = in_sizes[1] / N;

    float* h = (float*)d_ws;                 // N*DDIM floats (25.6 MB)
    const int grid = (N + ROWS - 1) / ROWS;

    sage_layer_wmma<<<grid, 256, 0, stream>>>(x, nbr, W1, b1, h,   N, deg);
    sage_layer_wmma<<<grid, 256, 0, stream>>>(h, nbr, W2, b2, out, N, deg);
}